// GAT_64939905516031
// MI455X (gfx1250) — compile-verified
//
#include <hip/hip_runtime.h>
#include <math.h>

typedef float v2f __attribute__((ext_vector_type(2)));
typedef float v8f __attribute__((ext_vector_type(8)));

#define LRELU_SLOPE 0.2f

// ---------- float <-> order-preserving uint (for atomic segment-max) ----------
__device__ __forceinline__ unsigned f2o(float f) {
  unsigned u = __float_as_uint(f);
  return (u & 0x80000000u) ? ~u : (u | 0x80000000u);
}
__device__ __forceinline__ float o2f(unsigned o) {
  unsigned u = (o & 0x80000000u) ? (o & 0x7fffffffu) : ~o;
  return __uint_as_float(u);
}
#define ORD_NEG_INF 0x007fffffu   // f2o(-inf)

// ---------------------------- fill kernels ----------------------------
__global__ void fill_f32(float* p, float v, int n) {
  int i = blockIdx.x * blockDim.x + threadIdx.x;
  int stride = gridDim.x * blockDim.x;
  for (; i < n; i += stride) p[i] = v;
}
__global__ void fill_u32(unsigned* p, unsigned v, int n) {
  int i = blockIdx.x * blockDim.x + threadIdx.x;
  int stride = gridDim.x * blockDim.x;
  for (; i < n; i += stride) p[i] = v;
}

// ------------------- fp32 WMMA GEMM: C[M,Ncols] = A[M,K] @ B[K,Ncols] -------------------
// Row-major. M % 16 == 0, Ncols % 16 == 0, K % 4 == 0.
// One wave (32 lanes) computes one 16x16 tile via V_WMMA_F32_16X16X4_F32.
// A 16x4 layout: lanes 0-15 -> M=l, v0=K0,v1=K1 ; lanes 16-31 -> M=l, v0=K2,v1=K3.
// B  4x16 layout: lanes 0-15 -> N=l, v0=K0 row, v1=K1 row ; lanes 16-31 -> K2/K3 rows.
// C/D 16x16 layout: VGPR r -> row M=r (lanes 0-15), M=r+8 (lanes 16-31), N=lane%16.
__global__ void gemm_wmma_f32(const float* __restrict__ A,
                              const float* __restrict__ B,
                              float* __restrict__ C,
                              int M, int Ncols, int K) {
  const int wavesPerBlock = blockDim.x >> 5;
  const int wave = threadIdx.x >> 5;
  const int lane = threadIdx.x & 31;
  const int half = lane >> 4;      // 0: lanes 0-15, 1: lanes 16-31
  const int l    = lane & 15;

  const int m0 = blockIdx.x * 16;
  const int nt = blockIdx.y * wavesPerBlock + wave;
  const int n0 = nt * 16;
  if (m0 >= M || n0 >= Ncols) return;

  v8f acc = {};
  const float* arow = A + (size_t)(m0 + l) * K + half * 2;
  for (int k = 0; k < K; k += 4) {
    v2f a, b;
    a.x = arow[k];
    a.y = arow[k + 1];
    const float* bp = B + (size_t)(k + half * 2) * Ncols + n0 + l;
    b.x = bp[0];
    b.y = bp[Ncols];
    acc = __builtin_amdgcn_wmma_f32_16x16x4_f32(false, a, false, b,
                                                (short)0, acc, false, false);
  }
#pragma unroll
  for (int r = 0; r < 8; ++r) {
    C[(size_t)(m0 + half * 8 + r) * Ncols + n0 + l] = acc[r];
  }
}

// ---------------- per-(node,head) attention logits: el = <feat, al>, er = <feat, ar> ----------------
__global__ void node_scores(const float* __restrict__ feat,
                            const float* __restrict__ al,
                            const float* __restrict__ ar,
                            float* __restrict__ el, float* __restrict__ er,
                            int N, int H, int D) {
  int i = blockIdx.x * blockDim.x + threadIdx.x;
  if (i >= N * H) return;
  int n = i / H, h = i - n * H;
  const float* f = feat + (size_t)n * H * D + h * D;
  const float* wl = al + h * D;
  const float* wr = ar + h * D;
  float sl = 0.f, sr = 0.f;
#pragma unroll 4
  for (int d = 0; d < D; ++d) {
    float v = f[d];
    sl += v * wl[d];
    sr += v * wr[d];
  }
  el[i] = sl;
  er[i] = sr;
}

// ---------------- edge pass 1: segment max of leaky_relu(el[src]+er[dst]) over dst ----------------
__global__ void edge_max(const int* __restrict__ src, const int* __restrict__ dst,
                         const float* __restrict__ el, const float* __restrict__ er,
                         unsigned* __restrict__ m_ord, int E, int H) {
  int i = blockIdx.x * blockDim.x + threadIdx.x;
  if (i >= E * H) return;
  int e = i / H, h = i - e * H;
  int d = dst[e];
  float sc = el[src[e] * H + h] + er[d * H + h];
  sc = sc > 0.f ? sc : LRELU_SLOPE * sc;
  atomicMax(m_ord + d * H + h, f2o(sc));
}

// ---------------- edge pass 2: ee = exp(score - m[dst]); s[dst] += ee ----------------
__global__ void edge_expsum(const int* __restrict__ src, const int* __restrict__ dst,
                            const float* __restrict__ el, const float* __restrict__ er,
                            const unsigned* __restrict__ m_ord,
                            float* __restrict__ ee, float* __restrict__ s,
                            int E, int H) {
  int i = blockIdx.x * blockDim.x + threadIdx.x;
  if (i >= E * H) return;
  int e = i / H, h = i - e * H;
  int d = dst[e];
  float sc = el[src[e] * H + h] + er[d * H + h];
  sc = sc > 0.f ? sc : LRELU_SLOPE * sc;
  float m = o2f(m_ord[d * H + h]);
  if (m < -1e30f) m = 0.f;                    // matches where(isfinite(m), m, 0)
  float v = __expf(sc - m);
  ee[i] = v;
  atomicAdd(s + d * H + h, v);
}

// ---------------- edge pass 3: acc[dst] += (ee / max(s[dst],1e-9)) * feat[src] ----------------
// One thread handles 4 consecutive f32 of a (head,dim) row; 16-byte coalesced gathers.
__global__ void edge_aggregate(const int* __restrict__ src, const int* __restrict__ dst,
                               const float* __restrict__ ee, const float* __restrict__ s,
                               const float* __restrict__ feat, float* __restrict__ acc,
                               int E, int H, int D) {
  const int chunks = (H * D) >> 2;            // float4 chunks per edge
  int tid = blockIdx.x * blockDim.x + threadIdx.x;
  if (tid >= E * chunks) return;
  int e = tid / chunks;
  int c = tid - e * chunks;
  int h = (c << 2) / D;
  int sN = src[e], dN = dst[e];
  float a = ee[(size_t)e * H + h] / fmaxf(s[(size_t)dN * H + h], 1e-9f);
  const float4 f = *(const float4*)(feat + (size_t)sN * H * D + (c << 2));
  float* o = acc + (size_t)dN * H * D + (c << 2);
  atomicAdd(o + 0, a * f.x);
  atomicAdd(o + 1, a * f.y);
  atomicAdd(o + 2, a * f.z);
  atomicAdd(o + 3, a * f.w);
}

// ---------------- layer-0 epilogue: h1[n,d] = (1/H) * sum_t sum_h elu(acc_t[n,h,d] + b[h,d]) ----------------
__global__ void epilogue0(const float* __restrict__ acc, const float* __restrict__ b,
                          float* __restrict__ h1, int N, int H, int D) {
  int i = blockIdx.x * blockDim.x + threadIdx.x;
  if (i >= N * D) return;
  int n = i / D, d = i - n * D;
  float sum = 0.f;
  for (int t = 0; t < 2; ++t) {
    const float* a = acc + ((size_t)t * N + n) * H * D + d;
    for (int h = 0; h < H; ++h) {
      float v = a[h * D] + b[h * D + d];
      sum += (v > 0.f) ? v : expm1f(v);
    }
  }
  h1[i] = sum / (float)H;
}

// ---------------- layer-1 epilogue (no activation, H=1): out = (acc0+b) + (acc1+b) ----------------
__global__ void epilogue1(const float* __restrict__ acc, const float* __restrict__ b,
                          float* __restrict__ out, int N, int D) {
  int i = blockIdx.x * blockDim.x + threadIdx.x;
  if (i >= N * D) return;
  int d = i - (i / D) * D;
  out[i] = (acc[i] + b[d]) + (acc[(size_t)N * D + i] + b[d]);
}

// ============================== host launch ==============================
extern "C" void kernel_launch(void* const* d_in, const int* in_sizes, int n_in,
                              void* d_out, int out_size, void* d_ws, size_t ws_size,
                              hipStream_t stream) {
  const int IN = 128, H0 = 4, D0 = 64, H1 = 1, D1 = 64;
  const int N = in_sizes[0] / IN;       // 50000
  const int E = in_sizes[9];            // 400000 per edge type

  const float* x   = (const float*)d_in[0];
  const float* W0  = (const float*)d_in[1];
  const float* al0 = (const float*)d_in[2];
  const float* ar0 = (const float*)d_in[3];
  const float* b0  = (const float*)d_in[4];
  const float* W1  = (const float*)d_in[5];
  const float* al1 = (const float*)d_in[6];
  const float* ar1 = (const float*)d_in[7];
  const float* b1  = (const float*)d_in[8];
  const int* srcT[2] = { (const int*)d_in[9],  (const int*)d_in[11] };
  const int* dstT[2] = { (const int*)d_in[10], (const int*)d_in[12] };
  float* out = (float*)d_out;

  // ---- scratch layout (floats). Layer-1 block aliases feat0 (dead by then). ----
  float* ws = (float*)d_ws;
  float*    feat0 = ws;                                 // N * H0*D0            (256/node)
  float*    el0   = feat0 + (size_t)N * H0 * D0;        // N*H0
  float*    er0   = el0 + (size_t)N * H0;               // N*H0
  unsigned* m0    = (unsigned*)(er0 + (size_t)N * H0);  // 2 * N*H0
  float*    s0    = (float*)(m0 + 2 * (size_t)N * H0);  // 2 * N*H0
  float*    ee0   = s0 + 2 * (size_t)N * H0;            // 2 * E*H0
  float*    acc0  = ee0 + 2 * (size_t)E * H0;           // 2 * N*H0*D0
  float*    h1    = acc0 + 2 * (size_t)N * H0 * D0;     // N*D0
  // layer-1 buffers overlaid on feat0's region (42.8 MB < 51.2 MB):
  float*    feat1 = feat0;                              // N*D1
  float*    el1   = feat1 + (size_t)N * D1;             // N
  float*    er1   = el1 + N;                            // N
  unsigned* m1    = (unsigned*)(er1 + N);               // 2*N
  float*    s1    = (float*)(m1 + 2 * (size_t)N);       // 2*N
  float*    ee1   = s1 + 2 * (size_t)N;                 // 2*E
  float*    acc1  = ee1 + 2 * (size_t)E;                // 2*N*D1

  const int TB = 256;
  const int fillBlocks = 2048;
  (void)ws_size; (void)n_in; (void)out_size;

  // ================= Layer 0 =================
  {
    // GEMM: feat0[N,256] = x[N,128] @ W0[128,256]
    dim3 grid(N / 16, (H0 * D0) / 16 / 4);  // 4 waves per block, each one 16x16 N-tile
    gemm_wmma_f32<<<grid, 128, 0, stream>>>(x, W0, feat0, N, H0 * D0, IN);

    node_scores<<<(N * H0 + TB - 1) / TB, TB, 0, stream>>>(feat0, al0, ar0, el0, er0, N, H0, D0);

    fill_u32<<<fillBlocks, TB, 0, stream>>>(m0, ORD_NEG_INF, 2 * N * H0);
    fill_f32<<<fillBlocks, TB, 0, stream>>>(s0, 0.f, 2 * N * H0);
    fill_f32<<<fillBlocks, TB, 0, stream>>>(acc0, 0.f, 2 * N * H0 * D0);

    for (int t = 0; t < 2; ++t) {
      unsigned* mT = m0 + (size_t)t * N * H0;
      float*    sT = s0 + (size_t)t * N * H0;
      float*    eeT = ee0 + (size_t)t * E * H0;
      float*    accT = acc0 + (size_t)t * N * H0 * D0;
      int eh = E * H0;
      edge_max<<<(eh + TB - 1) / TB, TB, 0, stream>>>(srcT[t], dstT[t], el0, er0, mT, E, H0);
      edge_expsum<<<(eh + TB - 1) / TB, TB, 0, stream>>>(srcT[t], dstT[t], el0, er0, mT, eeT, sT, E, H0);
      int total = E * ((H0 * D0) >> 2);
      edge_aggregate<<<(total + TB - 1) / TB, TB, 0, stream>>>(srcT[t], dstT[t], eeT, sT, feat0, accT, E, H0, D0);
    }

    epilogue0<<<(N * D0 + TB - 1) / TB, TB, 0, stream>>>(acc0, b0, h1, N, H0, D0);
  }

  // ================= Layer 1 =================
  {
    // GEMM: feat1[N,64] = h1[N,64] @ W1[64,64]
    dim3 grid(N / 16, 1);                   // 4 waves cover the 4 N-tiles
    gemm_wmma_f32<<<grid, 128, 0, stream>>>(h1, W1, feat1, N, H1 * D1, D0);

    node_scores<<<(N * H1 + TB - 1) / TB, TB, 0, stream>>>(feat1, al1, ar1, el1, er1, N, H1, D1);

    fill_u32<<<fillBlocks, TB, 0, stream>>>(m1, ORD_NEG_INF, 2 * N * H1);
    fill_f32<<<fillBlocks, TB, 0, stream>>>(s1, 0.f, 2 * N * H1);
    fill_f32<<<fillBlocks, TB, 0, stream>>>(acc1, 0.f, 2 * N * H1 * D1);

    for (int t = 0; t < 2; ++t) {
      unsigned* mT = m1 + (size_t)t * N * H1;
      float*    sT = s1 + (size_t)t * N * H1;
      float*    eeT = ee1 + (size_t)t * E * H1;
      float*    accT = acc1 + (size_t)t * N * H1 * D1;
      int eh = E * H1;
      edge_max<<<(eh + TB - 1) / TB, TB, 0, stream>>>(srcT[t], dstT[t], el1, er1, mT, E, H1);
      edge_expsum<<<(eh + TB - 1) / TB, TB, 0, stream>>>(srcT[t], dstT[t], el1, er1, mT, eeT, sT, E, H1);
      int total = E * ((H1 * D1) >> 2);
      edge_aggregate<<<(total + TB - 1) / TB, TB, 0, stream>>>(srcT[t], dstT[t], eeT, sT, feat1, accT, E, H1, D1);
    }

    epilogue1<<<(N * D1 + TB - 1) / TB, TB, 0, stream>>>(acc1, b1, out, N, D1);
  }
}